// SiT_Clf_57019985822468
// MI455X (gfx1250) — compile-verified
//
#include <hip/hip_runtime.h>
#include <math.h>

// ---------------- model constants ----------------
#define BB    8
#define NN    1024
#define DD    1024
#define HH    16
#define HD    64
#define LL    8
#define MLPD  4096
#define SCALE 0.125f          // HEAD^-0.5 = 64^-0.5

typedef float v2f __attribute__((ext_vector_type(2)));
typedef float v8f __attribute__((ext_vector_type(8)));
typedef int   v4i __attribute__((vector_size(4 * sizeof(int))));

__device__ __forceinline__ v8f wmma_f32(v2f a, v2f b, v8f c) {
  // V_WMMA_F32_16X16X4_F32 : D = A(16x4) * B(4x16) + C(16x16)
  return __builtin_amdgcn_wmma_f32_16x16x4_f32(false, a, false, b, (short)0, c,
                                               false, false);
}

__device__ __forceinline__ float silu_f(float x) {
  return x / (1.0f + expf(-x));
}
__device__ __forceinline__ float gelu_tanh(float x) {
  float x3 = x * x * x;
  return 0.5f * x * (1.0f + tanhf(0.7978845608028654f * (x + 0.044715f * x3)));
}

// ---------------- async global->LDS copy (CDNA5 ASYNCcnt path) -------------
#if defined(__HIP_DEVICE_COMPILE__) && \
    __has_builtin(__builtin_amdgcn_global_load_async_to_lds_b128)
#define HAVE_ASYNC_LDS 1
#else
#define HAVE_ASYNC_LDS 0
#endif

__device__ __forceinline__ void async_copy16(const float* __restrict__ g,
                                             float* l) {
#if HAVE_ASYNC_LDS
  __builtin_amdgcn_global_load_async_to_lds_b128(
      (__attribute__((address_space(1))) v4i*)g,
      (__attribute__((address_space(3))) v4i*)l, 0, 0);
#else
  *(float4*)l = *(const float4*)g;
#endif
}

#if HAVE_ASYNC_LDS
#if __has_builtin(__builtin_amdgcn_s_wait_asynccnt)
#define WAIT_ASYNC(n) __builtin_amdgcn_s_wait_asynccnt(n)
#else
#define WAIT_ASYNC(n) asm volatile("s_wait_asynccnt %0" ::"i"(n) : "memory")
#endif
#else
#define WAIT_ASYNC(n) ((void)0)
#endif

// ---------------- timestep embedding (t == 1 for all b) ----------------
// u = silu(t_freq @ t_w1 + t_b1)   [1024]
__global__ __launch_bounds__(256) void k_ts1(const float* __restrict__ w1,
                                             const float* __restrict__ b1,
                                             float* __restrict__ u) {
  int j = blockIdx.x * 256 + threadIdx.x;   // 0..1023
  float acc = 0.0f;
  for (int i = 0; i < 256; ++i) {
    int ii = (i < 128) ? i : (i - 128);
    float f = expf(-9.210340371976184f * (float)ii / 128.0f);  // freqs
    float tf = (i < 128) ? cosf(f) : sinf(f);
    acc += tf * w1[(size_t)i * DD + j];
  }
  acc += b1[j];
  u[j] = silu_f(acc);
}

// s = silu(u @ t_w2 + t_b2)  [1024]  (silu(c), the only thing mod needs)
__global__ __launch_bounds__(256) void k_ts2(const float* __restrict__ u,
                                             const float* __restrict__ w2,
                                             const float* __restrict__ b2,
                                             float* __restrict__ s) {
  int d = blockIdx.x * 256 + threadIdx.x;
  float acc = 0.0f;
  for (int j = 0; j < DD; ++j) acc += u[j] * w2[(size_t)j * DD + d];
  acc += b2[d];
  s[d] = silu_f(acc);
}

// mod[l][j] = silu(c) @ ada_w[l] + ada_b[l]   (batch-independent, all layers)
__global__ __launch_bounds__(256) void k_mod(const float* __restrict__ s,
                                             const float* __restrict__ aw,
                                             const float* __restrict__ ab,
                                             float* __restrict__ modv) {
  __shared__ float ssh[DD];
  int l = blockIdx.y;
  int j = blockIdx.x * 256 + threadIdx.x;     // 0..6143
  for (int i = threadIdx.x; i < DD; i += 256) ssh[i] = s[i];
  __syncthreads();
  float acc = 0.0f;
  const float* W = aw + (size_t)l * DD * (6 * DD) + j;
  for (int d = 0; d < DD; ++d) acc += ssh[d] * W[(size_t)d * (6 * DD)];
  modv[(size_t)l * (6 * DD) + j] = acc + ab[(size_t)l * (6 * DD) + j];
}

// ---------------- positional embed: h = x @ pos_w + pos_b ----------------
__global__ __launch_bounds__(256) void k_pos(const float* __restrict__ x,
                                             const float* __restrict__ pw,
                                             const float* __restrict__ pb,
                                             float* __restrict__ h) {
  int d = blockIdx.x * 256 + threadIdx.x;
  size_t row = blockIdx.y;
  const float* xr = x + row * 3;
  h[row * DD + d] =
      xr[0] * pw[d] + xr[1] * pw[DD + d] + xr[2] * pw[2 * DD + d] + pb[d];
}

// ---------------- LayerNorm + adaLN modulate ----------------
// xm = LN(h) * (1 + sc) + sh     (one row per wave, 8 rows per block)
__global__ __launch_bounds__(256) void k_lnmod(const float* __restrict__ h,
                                               const float* __restrict__ sc,
                                               const float* __restrict__ sh,
                                               float* __restrict__ xm) {
  int lane = threadIdx.x & 31;
  int w = threadIdx.x >> 5;
  size_t row = (size_t)blockIdx.x * 8 + w;
  const float* hr = h + row * DD;
  float vals[32];
  float s1 = 0.0f, s2 = 0.0f;
#pragma unroll
  for (int i = 0; i < 32; ++i) {
    float v = hr[lane + 32 * i];
    vals[i] = v; s1 += v; s2 += v * v;
  }
#pragma unroll
  for (int m = 16; m >= 1; m >>= 1) {
    s1 += __shfl_xor(s1, m, 32);
    s2 += __shfl_xor(s2, m, 32);
  }
  float mu = s1 * (1.0f / (float)DD);
  float var = s2 * (1.0f / (float)DD) - mu * mu;
  float r = rsqrtf(var + 1e-6f);
  float* o = xm + row * DD;
#pragma unroll
  for (int i = 0; i < 32; ++i) {
    int c = lane + 32 * i;
    o[c] = (vals[i] - mu) * r * (1.0f + sc[c]) + sh[c];
  }
}

// ---------------- fp32 WMMA GEMM: out = epilogue(A[MxK] @ W[KxN] + bias) ----
// Block tile 64x64, BK=16, 8 waves; wave (wm,wn) computes a 16x32 C strip.
// Double-buffered LDS tiles filled with global_load_async_to_lds_b128.
#define BM 64
#define BN 64
#define BK 16
#define ASTR 20   // As row stride: 80B (16B-aligned); 20*i mod 64 distinct, i<16
#define BSTR 68   // Bs row stride: 272B (16B-aligned); separates lane-half rows

enum { EP_BIAS = 0, EP_GELU = 1, EP_RESID = 2 };

template <int EP>
__global__ __launch_bounds__(256) void k_gemm(
    const float* __restrict__ A, const float* __restrict__ W,
    const float* __restrict__ bias, float* __restrict__ out,
    const float* __restrict__ gate, int M, int N, int K) {
  __shared__ __align__(16) float As[2][BM][ASTR];
  __shared__ __align__(16) float Bs[2][BK][BSTR];
  const int tid = threadIdx.x;
  const int lane = tid & 31;
  const int wid = tid >> 5;
  const int wm = wid & 3;         // M sub-tile: wm*16
  const int wn = wid >> 2;        // N sub-tile: wn*32
  const int m0 = blockIdx.y * BM;
  const int n0 = blockIdx.x * BN;
  const int lr = lane & 15;
  const int kh = (lane >> 4) << 1;    // 0 or 2 (K split across lane halves)

  // staging assignment: 256 threads x one 16B chunk for each tile
  const int ar = tid >> 2;            // A row 0..63
  const int ac = (tid & 3) << 2;      // A col {0,4,8,12}
  const int bk = tid >> 4;            // B row 0..15
  const int bn = (tid & 15) << 2;     // B col {0,4,...,60}

  v8f acc0 = {}; v8f acc1 = {};

  auto stage = [&](int buf, int kb) {
    async_copy16(A + (size_t)(m0 + ar) * K + kb + ac, &As[buf][ar][ac]);
    async_copy16(W + (size_t)(kb + bk) * N + n0 + bn, &Bs[buf][bk][bn]);
  };

  stage(0, 0);
  for (int kb = 0; kb < K; kb += BK) {
    const int buf = (kb >> 4) & 1;
    const bool more = (kb + BK) < K;
    if (more) {
      stage(buf ^ 1, kb + BK);
      WAIT_ASYNC(2);     // retire current buffer's 2 ops; prefetch in flight
    } else {
      WAIT_ASYNC(0);
    }
    __syncthreads();
#pragma unroll
    for (int kk = 0; kk < 4; ++kk) {
      int kc = kk * 4 + kh;
      v2f a = *(const v2f*)&As[buf][wm * 16 + lr][kc];
      v2f b0, b1;
      b0.x = Bs[buf][kc + 0][wn * 32 + lr];
      b0.y = Bs[buf][kc + 1][wn * 32 + lr];
      b1.x = Bs[buf][kc + 0][wn * 32 + 16 + lr];
      b1.y = Bs[buf][kc + 1][wn * 32 + 16 + lr];
      acc0 = wmma_f32(a, b0, acc0);
      acc1 = wmma_f32(a, b1, acc1);
    }
    __syncthreads();   // all waves done with buf before it is restaged
  }

  // epilogue: C element (vgpr i, lane) -> M=(lane>>4)*8+i, N=lane&15
  int rowb = m0 + wm * 16 + (lane >> 4) * 8;
#pragma unroll
  for (int t = 0; t < 2; ++t) {
    v8f acc = t ? acc1 : acc0;
    int col = n0 + wn * 32 + t * 16 + lr;
    float bs = bias[col];
    float gv = (EP == EP_RESID) ? gate[col] : 0.0f;
#pragma unroll
    for (int i = 0; i < 8; ++i) {
      float v = acc[i] + bs;
      size_t idx = (size_t)(rowb + i) * N + col;
      if (EP == EP_BIAS)      out[idx] = v;
      else if (EP == EP_GELU) out[idx] = gelu_tanh(v);
      else                    out[idx] = out[idx] + gv * v;   // gated residual
    }
  }
}

// ---------------- attention: one block per (b, head, 16-query tile) --------
// 4 waves. Phase1: S = Q K^T (WMMA, fragments straight from global since
// head-dim is contiguous). Phase2: masked softmax in LDS (XOR-swizzled, 64KB).
// Phase3: O = P V (WMMA, P fragments from LDS).
__device__ __forceinline__ int sidx(int m, int c) {
  return m * 1024 + ((((c >> 1) ^ m)) << 1) + (c & 1);   // pair-granular XOR swizzle
}

__global__ __launch_bounds__(128) void k_attn(const float* __restrict__ qkv,
                                              const unsigned char* __restrict__ pm,
                                              float* __restrict__ o) {
  __shared__ float S[16 * 1024];   // 64KB scores/probs for 16 queries
  const int tid = threadIdx.x;
  const int lane = tid & 31;
  const int w = tid >> 5;                       // wave 0..3
  const int qt = blockIdx.x & 63;
  const int hh = (blockIdx.x >> 6) & 15;
  const int bb = blockIdx.x >> 10;
  const int q0 = qt * 16;
  const size_t rs = 3 * DD;                     // qkv row stride
  const float* Qb = qkv + ((size_t)bb * NN) * rs + hh * HD;
  const float* Kb = Qb + DD;
  const float* Vb = Qb + 2 * DD;
  const int lr = lane & 15;
  const int kh = (lane >> 4) << 1;

  // preload the 16 Q fragments (A matrix 16x64, K-steps of 4)
  v2f qf[16];
#pragma unroll
  for (int ds = 0; ds < 16; ++ds)
    qf[ds] = *(const v2f*)(Qb + (size_t)(q0 + lr) * rs + ds * 4 + kh);

  bool qvs[8];
#pragma unroll
  for (int i = 0; i < 8; ++i)
    qvs[i] = pm[bb * NN + q0 + (lane >> 4) * 8 + i] != 0;

  // phase 1: scores; wave w covers key tiles w*16 .. w*16+15
  for (int t = 0; t < 16; ++t) {
    int kt0 = (w * 16 + t) * 16;
    v8f acc = {};
#pragma unroll
    for (int ds = 0; ds < 16; ++ds) {
      v2f bf = *(const v2f*)(Kb + (size_t)(kt0 + lr) * rs + ds * 4 + kh);
      acc = wmma_f32(qf[ds], bf, acc);
    }
    int kcol = kt0 + lr;
    bool kv = pm[bb * NN + kcol] != 0;
#pragma unroll
    for (int i = 0; i < 8; ++i) {
      int m = (lane >> 4) * 8 + i;
      float v = (kv && qvs[i]) ? acc[i] * SCALE : -1e9f;
      S[sidx(m, kcol)] = v;
    }
  }
  __syncthreads();

  // phase 2: softmax per row; 8 contiguous lanes per row
  {
    int r = tid >> 3, j = tid & 7;
    float mx = -3.0e38f;
    for (int c = j; c < NN; c += 8) mx = fmaxf(mx, S[sidx(r, c)]);
    mx = fmaxf(mx, __shfl_xor(mx, 1, 32));
    mx = fmaxf(mx, __shfl_xor(mx, 2, 32));
    mx = fmaxf(mx, __shfl_xor(mx, 4, 32));
    float sum = 0.0f;
    for (int c = j; c < NN; c += 8) {
      float p = __expf(S[sidx(r, c)] - mx);
      S[sidx(r, c)] = p; sum += p;
    }
    sum += __shfl_xor(sum, 1, 32);
    sum += __shfl_xor(sum, 2, 32);
    sum += __shfl_xor(sum, 4, 32);
    float inv = 1.0f / sum;
    for (int c = j; c < NN; c += 8) S[sidx(r, c)] *= inv;
  }
  __syncthreads();

  // phase 3: O = P @ V ; wave w owns head-dim cols [w*16, w*16+16)
  v8f acc = {};
  const float* Vc = Vb + w * 16 + lr;
  for (int ks = 0; ks < 256; ++ks) {
    int k0 = ks * 4 + kh;
    v2f pf = *(const v2f*)&S[sidx(lr, k0)];
    v2f vf;
    vf.x = Vc[(size_t)(k0 + 0) * rs];
    vf.y = Vc[(size_t)(k0 + 1) * rs];
    acc = wmma_f32(pf, vf, acc);
  }
  float* ob = o + ((size_t)bb * NN + q0) * DD + hh * HD + w * 16 + lr;
#pragma unroll
  for (int i = 0; i < 8; ++i)
    ob[(size_t)((lane >> 4) * 8 + i) * DD] = acc[i];
}

// ---------------- mean-pool + classifier ----------------
__global__ __launch_bounds__(256) void k_pool(const float* __restrict__ h,
                                              const float* __restrict__ cw,
                                              const float* __restrict__ cb,
                                              float* __restrict__ out) {
  __shared__ float red[256];
  int b = blockIdx.x, tid = threadIdx.x;
  float part = 0.0f;
  for (int d = tid; d < DD; d += 256) {
    const float* hp = h + (size_t)b * NN * DD + d;
    float colsum = 0.0f;
    for (int n = 0; n < NN; ++n) colsum += hp[(size_t)n * DD];
    part += colsum * cw[d];
  }
  red[tid] = part; __syncthreads();
  for (int st = 128; st > 0; st >>= 1) {
    if (tid < st) red[tid] += red[tid + st];
    __syncthreads();
  }
  if (tid == 0) out[b] = red[0] * (1.0f / (float)NN) + cb[0];
}

// ---------------- host orchestration ----------------
extern "C" void kernel_launch(void* const* d_in, const int* in_sizes, int n_in,
                              void* d_out, int out_size, void* d_ws,
                              size_t ws_size, hipStream_t stream) {
  const float* x      = (const float*)d_in[0];
  const unsigned char* pm = (const unsigned char*)d_in[1];  // bool mask
  const float* pos_w  = (const float*)d_in[2];
  const float* pos_b  = (const float*)d_in[3];
  const float* t_w1   = (const float*)d_in[4];
  const float* t_b1   = (const float*)d_in[5];
  const float* t_w2   = (const float*)d_in[6];
  const float* t_b2   = (const float*)d_in[7];
  const float* qkv_w  = (const float*)d_in[8];
  const float* qkv_b  = (const float*)d_in[9];
  const float* proj_w = (const float*)d_in[10];
  const float* proj_b = (const float*)d_in[11];
  const float* fc1_w  = (const float*)d_in[12];
  const float* fc1_b  = (const float*)d_in[13];
  const float* fc2_w  = (const float*)d_in[14];
  const float* fc2_b  = (const float*)d_in[15];
  const float* ada_w  = (const float*)d_in[16];
  const float* ada_b  = (const float*)d_in[17];
  const float* clf_w  = (const float*)d_in[18];
  const float* clf_b  = (const float*)d_in[19];

  float* ws   = (float*)d_ws;
  float* u    = ws;                          // 1024
  float* s    = ws + 1024;                   // 1024 (silu(c))
  float* modv = ws + 2048;                   // L*6144
  float* h    = ws + 65536;                  // 8192*1024
  float* xm   = h + (size_t)8192 * 1024;     // 8192*1024 (xm / attn out)
  float* big  = xm + (size_t)8192 * 1024;    // 8192*4096 (qkv / mlp hidden)

  const int M = BB * NN;  // 8192

  k_ts1<<<dim3(4), 256, 0, stream>>>(t_w1, t_b1, u);
  k_ts2<<<dim3(4), 256, 0, stream>>>(u, t_w2, t_b2, s);
  k_mod<<<dim3(24, LL), 256, 0, stream>>>(s, ada_w, ada_b, modv);
  k_pos<<<dim3(4, M), 256, 0, stream>>>(x, pos_w, pos_b, h);

  for (int l = 0; l < LL; ++l) {
    const float* md = modv + (size_t)l * 6 * DD;
    // attention branch
    k_lnmod<<<dim3(M / 8), 256, 0, stream>>>(h, md + DD, md, xm);   // sc_a, sh_a
    k_gemm<EP_BIAS><<<dim3(48, 128), 256, 0, stream>>>(
        xm, qkv_w + (size_t)l * DD * 3 * DD, qkv_b + (size_t)l * 3 * DD, big,
        nullptr, M, 3 * DD, DD);
    k_attn<<<dim3(BB * HH * (NN / 16)), 128, 0, stream>>>(big, pm, xm);
    k_gemm<EP_RESID><<<dim3(16, 128), 256, 0, stream>>>(
        xm, proj_w + (size_t)l * DD * DD, proj_b + (size_t)l * DD, h,
        md + 2 * DD, M, DD, DD);                                    // g_a
    // MLP branch
    k_lnmod<<<dim3(M / 8), 256, 0, stream>>>(h, md + 4 * DD, md + 3 * DD, xm);
    k_gemm<EP_GELU><<<dim3(64, 128), 256, 0, stream>>>(
        xm, fc1_w + (size_t)l * DD * MLPD, fc1_b + (size_t)l * MLPD, big,
        nullptr, M, MLPD, DD);
    k_gemm<EP_RESID><<<dim3(16, 128), 256, 0, stream>>>(
        big, fc2_w + (size_t)l * MLPD * DD, fc2_b + (size_t)l * DD, h,
        md + 5 * DD, M, DD, MLPD);                                  // g_m
  }

  k_pool<<<dim3(BB), 256, 0, stream>>>(h, clf_w, clf_b, (float*)d_out);
}